// DynamicGraphConstruction_49091476193878
// MI455X (gfx1250) — compile-verified
//
#include <hip/hip_runtime.h>
#include <math.h>

typedef float v2f __attribute__((ext_vector_type(2)));
typedef float v8f __attribute__((ext_vector_type(8)));

#define DIM    64
#define KNN    8
#define NBATCH 8
#define BN_EPS 1e-5f

// ---------------------------------------------------------------------------
// Workspace layout (floats):
//   [0, 2*NB)          : partial sums (NB for sum, NB for sumsq; reused for sum_w)
//   [2*NB, 2*NB+8)     : scalars {mean, scale, beta, normfactor}
//   then int bstart[8], int bend[8]
//   then float sqd[M]
// d_out layout (floats): [0,Nk)=graph row0 (src idx), [Nk,2Nk)=graph row1 (nn idx),
//                        [2Nk,3Nk)=w  (also used as lik scratch before normalize)
// ---------------------------------------------------------------------------

__global__ void k_init(int* bstart, int* bend, int M) {
  int t = threadIdx.x;
  if (t < NBATCH) { bstart[t] = M; bend[t] = 0; }
}

__global__ void k_pre(const float* __restrict__ dst, const int* __restrict__ dstb,
                      float* __restrict__ sqd, int* bstart, int* bend, int M) {
  int i = blockIdx.x * blockDim.x + threadIdx.x;
  if (i >= M) return;
  const float4* r = (const float4*)(dst + (size_t)i * DIM);
  float s = 0.f;
#pragma unroll
  for (int t = 0; t < DIM / 4; ++t) {
    float4 v = r[t];
    s += v.x * v.x + v.y * v.y + v.z * v.z + v.w * v.w;
  }
  sqd[i] = s;
  int b = dstb[i];
  atomicMin(&bstart[b], i);
  atomicMax(&bend[b], i + 1);
}

// Each 256-thread block = 8 waves; wave w owns src rows [blk*128+w*16, +16).
// Loop over 16-wide dst tiles restricted to the block's batch range.
__global__ void __launch_bounds__(256) k_knn(
    const float* __restrict__ src, const float* __restrict__ dst,
    const int* __restrict__ srcb, const int* __restrict__ dstb,
    const float* __restrict__ sqd, const int* __restrict__ bstart,
    const int* __restrict__ bend, float* __restrict__ outG0,
    float* __restrict__ outG1, int N) {
  __shared__ float Bt[16][68];       // dst tile, padded (stride 68 -> no conflicts)
  __shared__ float Ct[8][16 * 17];   // per-wave transpose / merge scratch
  __shared__ float sqd_s[16];
  __shared__ int   db_s[16];

  const int lane = threadIdx.x & 31;
  const int wave = threadIdx.x >> 5;
  const int m    = lane & 15;        // row (A) / col (B) within tile
  const int hi   = lane >> 4;
  const int kq   = hi * 2;           // K sub-offset per WMMA layout
  const int row0 = blockIdx.x * 128 + wave * 16;

  // A fragment: 16 rows x 64 K, resident for whole loop (v2f per K-step of 4)
  v2f a[16];
  const float* sr = src + (size_t)(row0 + m) * DIM + kq;
#pragma unroll
  for (int kk = 0; kk < 16; ++kk) a[kk] = *(const v2f*)(sr + kk * 4);
  const int sb = srcb[row0 + m];

  // Block-uniform dst range from sorted batches
  const int br0 = blockIdx.x * 128;
  int minb = srcb[br0];
  int maxb = srcb[br0 + 127];
  int start = N, end = 0;
  for (int b = minb; b <= maxb; ++b) {
    int s0 = bstart[b], e0 = bend[b];
    start = s0 < start ? s0 : start;
    end   = e0 > end   ? e0 : end;
  }
  const int j0 = start & ~15;

  float kd[KNN]; int ki[KNN];
#pragma unroll
  for (int t = 0; t < KNN; ++t) { kd[t] = __builtin_inff(); ki[t] = 0; }

  float* ct = &Ct[wave][0];
  const int rr = m;            // row this lane scans
  const int cbase = hi * 8;    // its 8 columns

  for (int j = j0; j < end; j += 16) {
    __syncthreads();
    { // cooperative tile load: 16 rows x 64 floats = 256 float4
      int t = threadIdx.x;
      int nr = t >> 4, c4 = (t & 15) * 4;
      *(float4*)&Bt[nr][c4] = *(const float4*)(dst + (size_t)(j + nr) * DIM + c4);
      if (t < 16) { sqd_s[t] = sqd[j + t]; db_s[t] = dstb[j + t]; }
    }
    __syncthreads();

    // C(16x16) = A(16x64) * B(64x16) via 16 chained fp32 WMMAs
    v8f c = {};
    const float* bl = &Bt[m][kq];
#pragma unroll
    for (int kk = 0; kk < 16; ++kk) {
      v2f bf = *(const v2f*)(bl + kk * 4);
      c = __builtin_amdgcn_wmma_f32_16x16x4_f32(false, a[kk], false, bf,
                                                (short)0, c, false, false);
    }

    // Transpose C fragment through per-wave LDS (stride 17)
#pragma unroll
    for (int r = 0; r < 8; ++r) ct[(r + 8 * hi) * 17 + m] = c[r];

    // Streaming top-8: key = ||d||^2 - 2*dot (||s||^2 constant per row)
#pragma unroll
    for (int c8 = 0; c8 < 8; ++c8) {
      int col = cbase + c8;
      float key = sqd_s[col] - 2.0f * ct[rr * 17 + col];
      if ((db_s[col] == sb) && (key < kd[KNN - 1])) {
        kd[KNN - 1] = key; ki[KNN - 1] = j + col;
#pragma unroll
        for (int t = KNN - 1; t > 0; --t) {
          float d0 = kd[t - 1], d1 = kd[t];
          int   i0 = ki[t - 1], i1 = ki[t];
          bool sw = d1 < d0;
          kd[t - 1] = sw ? d1 : d0;  kd[t] = sw ? d0 : d1;
          ki[t - 1] = sw ? i1 : i0;  ki[t] = sw ? i0 : i1;
        }
      }
    }
  }
  __syncthreads();

  // Merge the two lanes per row (each holds a sorted 8-list)
  if (lane >= 16) {
#pragma unroll
    for (int t = 0; t < KNN; ++t) {
      ct[(lane - 16) * 16 + t]     = kd[t];
      ct[(lane - 16) * 16 + 8 + t] = __int_as_float(ki[t]);
    }
  }
  if (lane < 16) {
    float bd[KNN]; int bi[KNN];
#pragma unroll
    for (int t = 0; t < KNN; ++t) {
      bd[t] = ct[lane * 16 + t];
      bi[t] = __float_as_int(ct[lane * 16 + 8 + t]);
    }
    int row = row0 + lane;
#pragma unroll
    for (int s = 0; s < KNN; ++s) {
      bool ta = kd[0] <= bd[0];
      int  oi = ta ? ki[0] : bi[0];
#pragma unroll
      for (int t = 0; t < KNN - 1; ++t) {
        kd[t] = ta ? kd[t + 1] : kd[t];
        ki[t] = ta ? ki[t + 1] : ki[t];
        bd[t] = ta ? bd[t] : bd[t + 1];
        bi[t] = ta ? bi[t] : bi[t + 1];
      }
      kd[KNN - 1] = ta ? __builtin_inff() : kd[KNN - 1];
      bd[KNN - 1] = ta ? bd[KNN - 1] : __builtin_inff();
      int e = row * KNN + s;
      outG0[e] = (float)row;
      outG1[e] = (float)oi;
    }
  }
}

// likelihood per edge + deterministic per-block partial sums (sum, sumsq)
__global__ void __launch_bounds__(256) k_lik(
    const float* __restrict__ src, const float* __restrict__ dst,
    const float* __restrict__ outG1, float* __restrict__ lik,
    float* __restrict__ part) {
  __shared__ float red[256];
  int e = blockIdx.x * 256 + threadIdx.x;
  int r = e >> 3;
  int nn = (int)outG1[e];
  const float4* s4 = (const float4*)(src + (size_t)r * DIM);
  const float4* d4 = (const float4*)(dst + (size_t)nn * DIM);
  float acc = 0.f;
#pragma unroll
  for (int t = 0; t < DIM / 4; ++t) {
    float4 x = s4[t], y = d4[t];
    acc += x.x * y.x + x.y * y.y + x.z * y.z + x.w * y.w;
  }
  lik[e] = acc;
  int t = threadIdx.x;
  red[t] = acc; __syncthreads();
  for (int s = 128; s > 0; s >>= 1) { if (t < s) red[t] += red[t + s]; __syncthreads(); }
  if (t == 0) part[blockIdx.x] = red[0];
  __syncthreads();
  red[t] = acc * acc; __syncthreads();
  for (int s = 128; s > 0; s >>= 1) { if (t < s) red[t] += red[t + s]; __syncthreads(); }
  if (t == 0) part[gridDim.x + blockIdx.x] = red[0];
}

__global__ void k_stats(const float* __restrict__ part, const float* __restrict__ gamma,
                        const float* __restrict__ beta, float* __restrict__ scal,
                        int NB, int Nk) {
  __shared__ float red[256];
  int t = threadIdx.x;
  float s = 0.f;
  for (int i = t; i < NB; i += 256) s += part[i];
  red[t] = s; __syncthreads();
  for (int k = 128; k > 0; k >>= 1) { if (t < k) red[t] += red[t + k]; __syncthreads(); }
  float total = red[0];
  __syncthreads();
  float q = 0.f;
  for (int i = t; i < NB; i += 256) q += part[NB + i];
  red[t] = q; __syncthreads();
  for (int k = 128; k > 0; k >>= 1) { if (t < k) red[t] += red[t + k]; __syncthreads(); }
  if (t == 0) {
    float mean = total / (float)Nk;
    float var  = red[0] / (float)Nk - mean * mean;
    scal[0] = mean;
    scal[1] = rsqrtf(var + BN_EPS) * gamma[0];
    scal[2] = beta[0];
  }
}

__global__ void __launch_bounds__(256) k_sig(float* __restrict__ wv,
                                             const float* __restrict__ scal,
                                             float* __restrict__ part) {
  __shared__ float red[256];
  int e = blockIdx.x * 256 + threadIdx.x;
  float x = (wv[e] - scal[0]) * scal[1] + scal[2];
  float w = 1.0f / (1.0f + expf(-x));
  wv[e] = w;
  int t = threadIdx.x;
  red[t] = w; __syncthreads();
  for (int s = 128; s > 0; s >>= 1) { if (t < s) red[t] += red[t + s]; __syncthreads(); }
  if (t == 0) part[blockIdx.x] = red[0];
}

__global__ void k_wmean(const float* __restrict__ part, float* __restrict__ scal,
                        int NB, int Nk) {
  __shared__ float red[256];
  int t = threadIdx.x;
  float s = 0.f;
  for (int i = t; i < NB; i += 256) s += part[i];
  red[t] = s; __syncthreads();
  for (int k = 128; k > 0; k >>= 1) { if (t < k) red[t] += red[t + k]; __syncthreads(); }
  if (t == 0) scal[3] = (float)Nk / red[0];
}

__global__ void k_scale(float* __restrict__ wv, const float* __restrict__ scal) {
  int e = blockIdx.x * 256 + threadIdx.x;
  wv[e] *= scal[3];
}

extern "C" void kernel_launch(void* const* d_in, const int* in_sizes, int n_in,
                              void* d_out, int out_size, void* d_ws, size_t ws_size,
                              hipStream_t stream) {
  const float* src   = (const float*)d_in[0];
  const float* dst   = (const float*)d_in[1];
  const int*   srcb  = (const int*)d_in[2];
  const int*   dstb  = (const int*)d_in[3];
  const float* gamma = (const float*)d_in[4];
  const float* beta  = (const float*)d_in[5];

  const int N  = in_sizes[0] / DIM;  // 16384
  const int M  = in_sizes[1] / DIM;  // 16384
  const int Nk = N * KNN;            // 131072
  const int NB = Nk / 256;           // 512

  float* wsf    = (float*)d_ws;
  float* part   = wsf;                       // 2*NB floats
  float* scal   = wsf + 2 * NB;              // 8 floats
  int*   bst    = (int*)(wsf + 2 * NB + 8);  // 8 ints
  int*   ben    = bst + NBATCH;              // 8 ints
  float* sqd    = (float*)(ben + NBATCH);    // M floats

  float* out   = (float*)d_out;
  float* outG0 = out;            // graph row 0 (src indices, as float)
  float* outG1 = out + Nk;       // graph row 1 (nn indices, as float)
  float* wv    = out + 2 * Nk;   // w (lik scratch first)

  k_init<<<1, 32, 0, stream>>>(bst, ben, M);
  k_pre<<<(M + 255) / 256, 256, 0, stream>>>(dst, dstb, sqd, bst, ben, M);
  k_knn<<<N / 128, 256, 0, stream>>>(src, dst, srcb, dstb, sqd, bst, ben,
                                     outG0, outG1, N);
  k_lik<<<NB, 256, 0, stream>>>(src, dst, outG1, wv, part);
  k_stats<<<1, 256, 0, stream>>>(part, gamma, beta, scal, NB, Nk);
  k_sig<<<NB, 256, 0, stream>>>(wv, scal, part);
  k_wmean<<<1, 256, 0, stream>>>(part, scal, NB, Nk);
  k_scale<<<NB, 256, 0, stream>>>(wv, scal);
}